// Qubits3Model_20366734917774
// MI455X (gfx1250) — compile-verified
//
#include <hip/hip_runtime.h>
#include <math.h>

// ---------------------------------------------------------------------------
// Qubits3 fidelity: two independent chains of 399 sequential 2x2 complex
// unitaries (analytic expm of the sparse block Hamiltonian), blocked scan:
//   - 32 lanes: 13-step local chunk products in f64 (VALU)
//   - log-depth combine tree on 4x4 real embeddings via V_WMMA_F32_16X16X4_F32
//     (4 independent 4x4 products per WMMA, block-diagonal packing)
//   - lane 0: fidelity formula in f64, f32 store
// Latency-bound problem (~50 KFLOPs total) -> single wave, minimum depth.
// ---------------------------------------------------------------------------

typedef __attribute__((ext_vector_type(2))) float v2f;
typedef __attribute__((ext_vector_type(8))) float v8f;

#define QN     399
#define CHUNK  13          // ceil(399/32)
#define TWO_PI 6.28318530717958647692

__global__ __launch_bounds__(32)
void qubits3_kernel(const double* __restrict__ Tp,
                    const double* __restrict__ psi,    // complex128: interleaved re,im (6 pairs)
                    const double* __restrict__ thetap,
                    const double* __restrict__ phi,
                    float* __restrict__ out)
{
    __shared__ double sphi[QN];
    __shared__ float  bufA[64 * 16];   // [chain*32 + idx] 4x4 f32 row-major
    __shared__ float  bufB[64 * 16];

    const int lane = threadIdx.x;      // 32 threads, 1 wave (wave32)

    // ---- cooperative, coalesced load of phi into LDS --------------------
    for (int i = lane; i < QN; i += 32) sphi[i] = phi[i];
    __syncthreads();

    // ---- per-lane f64 chunk product (both chains) -----------------------
    const double dt = Tp[0] / (double)QN;
    const double a0 = 0.5 * dt;                    // block (0,1): g = 1
    const double a1 = 0.5 * 1.41421356237309504880 * dt; // block (2,3): g = sqrt(2)
    const double C0 = cos(a0), S0 = sin(a0);
    const double C1 = cos(a1), S1 = sin(a1);

    // chain 0 product P, chain 1 product Q; start = identity
    double p00r = 1, p00i = 0, p01r = 0, p01i = 0, p10r = 0, p10i = 0, p11r = 1, p11i = 0;
    double q00r = 1, q00i = 0, q01r = 0, q01i = 0, q10r = 0, q10i = 0, q11r = 1, q11i = 0;

    const int kbeg = lane * CHUNK;
    const int kend = min(QN, kbeg + CHUNK);
    for (int k = kbeg; k < kend; ++k) {
        double sph, cph;
        sincos(sphi[k], &sph, &cph);
        // M = [[C, (S*sin, -S*cos)], [(-S*sin, -S*cos), C]];  P <- M*P
        {
            const double m01r = S0 * sph, m01i = -S0 * cph;
            const double m10r = -S0 * sph, m10i = -S0 * cph;
            const double n00r = C0 * p00r + m01r * p10r - m01i * p10i;
            const double n00i = C0 * p00i + m01r * p10i + m01i * p10r;
            const double n01r = C0 * p01r + m01r * p11r - m01i * p11i;
            const double n01i = C0 * p01i + m01r * p11i + m01i * p11r;
            const double n10r = C0 * p10r + m10r * p00r - m10i * p00i;
            const double n10i = C0 * p10i + m10r * p00i + m10i * p00r;
            const double n11r = C0 * p11r + m10r * p01r - m10i * p01i;
            const double n11i = C0 * p11i + m10r * p01i + m10i * p01r;
            p00r = n00r; p00i = n00i; p01r = n01r; p01i = n01i;
            p10r = n10r; p10i = n10i; p11r = n11r; p11i = n11i;
        }
        {
            const double m01r = S1 * sph, m01i = -S1 * cph;
            const double m10r = -S1 * sph, m10i = -S1 * cph;
            const double n00r = C1 * q00r + m01r * q10r - m01i * q10i;
            const double n00i = C1 * q00i + m01r * q10i + m01i * q10r;
            const double n01r = C1 * q01r + m01r * q11r - m01i * q11i;
            const double n01i = C1 * q01i + m01r * q11i + m01i * q11r;
            const double n10r = C1 * q10r + m10r * q00r - m10i * q00i;
            const double n10i = C1 * q10i + m10r * q00i + m10i * q00r;
            const double n11r = C1 * q11r + m10r * q01r - m10i * q01i;
            const double n11i = C1 * q11i + m10r * q01i + m10i * q01r;
            q00r = n00r; q00i = n00i; q01r = n01r; q01i = n01i;
            q10r = n10r; q10i = n10i; q11r = n11r; q11i = n11i;
        }
    }

    // ---- embed complex 2x2 -> real 4x4 [[Re,-Im],[Im,Re]], store f32 ----
    {
        float* m = &bufA[(0 * 32 + lane) * 16];
        m[0] = (float)p00r;  m[1] = (float)p01r;  m[2] = (float)(-p00i); m[3] = (float)(-p01i);
        m[4] = (float)p10r;  m[5] = (float)p11r;  m[6] = (float)(-p10i); m[7] = (float)(-p11i);
        m[8] = (float)p00i;  m[9] = (float)p01i;  m[10] = (float)p00r;  m[11] = (float)p01r;
        m[12] = (float)p10i; m[13] = (float)p11i; m[14] = (float)p10r;  m[15] = (float)p11r;
        m = &bufA[(1 * 32 + lane) * 16];
        m[0] = (float)q00r;  m[1] = (float)q01r;  m[2] = (float)(-q00i); m[3] = (float)(-q01i);
        m[4] = (float)q10r;  m[5] = (float)q11r;  m[6] = (float)(-q10i); m[7] = (float)(-q11i);
        m[8] = (float)q00i;  m[9] = (float)q01i;  m[10] = (float)q00r;  m[11] = (float)q01r;
        m[12] = (float)q10i; m[13] = (float)q11i; m[14] = (float)q10r;  m[15] = (float)q11r;
    }
    __syncthreads();

    // ---- combine tree: ordered pairwise products, 4 pairs per WMMA ------
    // order: new[t] = old[2t+1] * old[2t]  (later steps multiply on the left)
    float* src = bufA;
    float* dst = bufB;
    for (int n = 32; n > 1; n >>= 1) {
        const int half_n = n >> 1;       // pairs per chain
        const int totalPairs = n;        // 2 chains
        const int numGroups = (totalPairs + 3) >> 2;
        for (int g = 0; g < numGroups; ++g) {
            const int li = lane & 15;
            const int kbase = (lane >> 4) << 1;     // K pair {0,1} or {2,3}
            const int slot = li >> 2;               // which 4x4 in this WMMA
            const int rc = li & 3;                  // A: local row; B: local col
            int q = g * 4 + slot;
            if (q >= totalPairs) q = 0;             // clamp (output gated below)
            const int chain = (q >= half_n) ? 1 : 0;
            const int t = q - chain * half_n;
            const float* Am = src + (chain * 32 + 2 * t + 1) * 16;  // left
            const float* Bm = src + (chain * 32 + 2 * t) * 16;      // right
            // A layout: lane=row M, vgpr v = K=kbase+v   (ISA 16x4 f32 table)
            v2f av; av.x = Am[rc * 4 + kbase]; av.y = Am[rc * 4 + kbase + 1];
            // B layout: lane=col N, vgpr v = K=kbase+v
            v2f bv; bv.x = Bm[kbase * 4 + rc]; bv.y = Bm[(kbase + 1) * 4 + rc];
            v8f cv = {};
#if __has_builtin(__builtin_amdgcn_wmma_f32_16x16x4_f32)
            v8f dvec = __builtin_amdgcn_wmma_f32_16x16x4_f32(
                false, av, false, bv, (short)0, cv, false, false);
#else
            v8f dvec = cv; // fallback marker; scalar path below does real work
            {
                for (int qq = lane; qq < totalPairs; qq += 32) {
                    if ((qq >> 2) != g) continue;
                    const int ch2 = (qq >= half_n) ? 1 : 0;
                    const int t2 = qq - ch2 * half_n;
                    const float* A2 = src + (ch2 * 32 + 2 * t2 + 1) * 16;
                    const float* B2 = src + (ch2 * 32 + 2 * t2) * 16;
                    float* D2 = dst + (ch2 * 32 + t2) * 16;
                    for (int r = 0; r < 4; ++r)
                        for (int c = 0; c < 4; ++c) {
                            float acc = 0.f;
                            for (int k = 0; k < 4; ++k) acc += A2[r * 4 + k] * B2[k * 4 + c];
                            D2[r * 4 + c] = acc;
                        }
                }
            }
#endif
#if __has_builtin(__builtin_amdgcn_wmma_f32_16x16x4_f32)
            // Extract diagonal 4x4 blocks of D (16x16 f32 C/D layout):
            //   M<8 : lane=N, vgpr=M ; M>=8 : lane=N+16, vgpr=M-8
            int blk = -1, cc = 0, vb = 0;
            if (lane < 4)                    { blk = 0; cc = lane;      vb = 0; }
            else if (lane < 8)               { blk = 1; cc = lane - 4;  vb = 4; }
            else if (lane >= 24 && lane < 28){ blk = 2; cc = lane - 24; vb = 0; }
            else if (lane >= 28)             { blk = 3; cc = lane - 28; vb = 4; }
            if (blk >= 0) {
                const int qo = g * 4 + blk;
                if (qo < totalPairs) {
                    const int cho = (qo >= half_n) ? 1 : 0;
                    const int to = qo - cho * half_n;
                    float* Dm = dst + (cho * 32 + to) * 16;
                    const float e0 = vb ? dvec[4] : dvec[0];
                    const float e1 = vb ? dvec[5] : dvec[1];
                    const float e2 = vb ? dvec[6] : dvec[2];
                    const float e3 = vb ? dvec[7] : dvec[3];
                    Dm[0 * 4 + cc] = e0;
                    Dm[1 * 4 + cc] = e1;
                    Dm[2 * 4 + cc] = e2;
                    Dm[3 * 4 + cc] = e3;
                }
            }
#endif
        }
        __syncthreads();
        float* tmp = src; src = dst; dst = tmp;
    }

    // ---- lane 0: fidelity in f64, f32 store -----------------------------
    if (lane == 0) {
        const float* R0 = src + (0 * 32 + 0) * 16;   // chain 0 total (4x4 embed)
        const float* R1 = src + (1 * 32 + 0) * 16;   // chain 1 total
        // U[0][0] = R[0][0] + i R[2][0]; U[0][1] = R[0][1] + i R[2][1]
        const double u00r = (double)R0[0], u00i = (double)R0[8];
        const double u01r = (double)R0[1], u01i = (double)R0[9];
        const double v00r = (double)R1[0], v00i = (double)R1[8];
        const double v01r = (double)R1[1], v01i = (double)R1[9];

        const double z0r = psi[0], z0i = psi[1];
        const double z1r = psi[2], z1i = psi[3];
        const double z2r = psi[4], z2i = psi[5];
        const double z3r = psi[6], z3i = psi[7];

        // psi_f[0] = U00*z0 + U01*z1 ; psi_f[2] = V00*z2 + V01*z3
        const double f0r = u00r * z0r - u00i * z0i + u01r * z1r - u01i * z1i;
        const double f0i = u00r * z0i + u00i * z0r + u01r * z1i + u01i * z1r;
        const double f2r = v00r * z2r - v00i * z2i + v01r * z3r - v01i * z3i;
        const double f2i = v00r * z2i + v00i * z2r + v01r * z3i + v01i * z3r;

        const double th = thetap[0];
        double s1, c1, s2, c2, s3, c3;
        sincos(th, &s1, &c1);
        sincos(2.0 * th, &s2, &c2);
        sincos(3.0 * th, &s3, &c3);
        // a001 = e^{-i th} * f0 ; a011 = e^{-2i th} * f2 ; a111 = -e^{-3i th} * f2
        const double a1r = c1 * f0r + s1 * f0i, a1i = c1 * f0i - s1 * f0r;
        const double a2r = c2 * f2r + s2 * f2i, a2i = c2 * f2i - s2 * f2r;
        const double a3r = -(c3 * f2r + s3 * f2i), a3i = -(c3 * f2i - s3 * f2r);

        const double Sr = 1.0 + 3.0 * a1r + 3.0 * a2r + a3r;
        const double Si = 3.0 * a1i + 3.0 * a2i + a3i;
        const double m1 = a1r * a1r + a1i * a1i;
        const double m2 = a2r * a2r + a2i * a2i;
        const double m3 = a3r * a3r + a3i * a3i;

        double F = (Sr * Sr + Si * Si + 1.0 + 3.0 * m1 + 3.0 * m2 + m3) / 72.0;
        F = fmin(fmax(F, 0.0), 1.0);
        out[0] = (float)F;
    }
}

extern "C" void kernel_launch(void* const* d_in, const int* in_sizes, int n_in,
                              void* d_out, int out_size, void* d_ws, size_t ws_size,
                              hipStream_t stream) {
    (void)in_sizes; (void)n_in; (void)d_ws; (void)ws_size; (void)out_size;
    const double* T     = (const double*)d_in[0];
    const double* psi   = (const double*)d_in[1];  // complex128 interleaved
    const double* theta = (const double*)d_in[2];
    const double* phi   = (const double*)d_in[3];
    qubits3_kernel<<<1, 32, 0, stream>>>(T, psi, theta, phi, (float*)d_out);
}